// GNNGate_11879879544433
// MI455X (gfx1250) — compile-verified
//
#include <hip/hip_runtime.h>
#include <math.h>

#define N_NODES   100000
#define N_EDGES   3200000
#define E_TOT     (N_EDGES + N_NODES)   // edges + self loops
#define DIM       256
#define HEADS     4
#define OUTC      4
#define F         16                    // HEADS*OUTC
#define NEG_SLOPE 0.2f
#define LN_EPS    1e-5f
#define NTILES    (N_NODES / 16)        // 6250 exactly

typedef float v2f __attribute__((ext_vector_type(2)));
typedef float v8f __attribute__((ext_vector_type(8)));

// ---------------------------------------------------------------------------
// Kernel 1: h = x @ W via V_WMMA_F32_16X16X4_F32 (fp32 tensor path), fused
// with the a_s / a_d attention einsums (cross-lane reduce over 4 channels).
// One wave32 computes one 16-node x 16-channel tile; K=256 in steps of 4.
// ---------------------------------------------------------------------------
__global__ __launch_bounds__(256) void gemm_att_kernel(
    const float* __restrict__ x, const float* __restrict__ W,
    const float* __restrict__ att_src, const float* __restrict__ att_dst,
    float* __restrict__ h, float* __restrict__ a_s, float* __restrict__ a_d)
{
    __shared__ float Ws[DIM * F];                 // 16 KB: W staged once/block
    for (int i = threadIdx.x; i < DIM * F; i += 256) Ws[i] = W[i];
    __syncthreads();

    const int wid  = threadIdx.x >> 5;
    const int lane = threadIdx.x & 31;
    const int tile = blockIdx.x * 8 + wid;        // 8 waves / block
    if (tile >= NTILES) return;                   // uniform per wave -> EXEC all-1s below

    const int half = lane >> 4;                   // 0 or 1 (K sub-pair select)
    const int mn   = lane & 15;                   // A-row / B-col index
    const int row0 = tile * 16;

    // A: lane holds x[row0+mn][k0 + 2*half + {0,1}]  (8B-aligned v2f load)
    const float* xrow = x + (size_t)(row0 + mn) * DIM + 2 * half;
    // B: lane holds W[k0 + 2*half + {0,1}][mn] from LDS
    const float* wcol = Ws + (2 * half) * F + mn;

    v8f acc = {0.f, 0.f, 0.f, 0.f, 0.f, 0.f, 0.f, 0.f};
    #pragma unroll 8
    for (int k0 = 0; k0 < DIM; k0 += 4) {
        v2f a = *(const v2f*)(xrow + k0);
        v2f b;
        b.x = wcol[(k0 + 0) * F];
        b.y = wcol[(k0 + 1) * F];
        acc = __builtin_amdgcn_wmma_f32_16x16x4_f32(
            /*neg_a=*/false, a, /*neg_b=*/false, b,
            /*c_mod=*/(short)0, acc, /*reuse_a=*/false, /*reuse_b=*/false);
    }

    // C/D layout: VGPR r -> row (r + 8*half), col mn.
    const float asw = att_src[mn];                // flat [H*OUT] = [16]
    const float adw = att_dst[mn];
    #pragma unroll
    for (int r = 0; r < 8; ++r) {
        const int row = row0 + r + 8 * half;
        h[(size_t)row * F + mn] = acc[r];         // coalesced 64B row stores
        // einsum over the 4 channels of this head: lanes {4k..4k+3} hold them
        float ts = acc[r] * asw;
        ts += __shfl_xor(ts, 1, 32);
        ts += __shfl_xor(ts, 2, 32);
        float td = acc[r] * adw;
        td += __shfl_xor(td, 1, 32);
        td += __shfl_xor(td, 2, 32);
        if ((mn & 3) == 0) {
            a_s[row * HEADS + (mn >> 2)] = ts;
            a_d[row * HEADS + (mn >> 2)] = td;
        }
    }
}

// ---------------------------------------------------------------------------
// Workspace init (d_ws is poisoned; must re-init every call)
// ---------------------------------------------------------------------------
__global__ void init_kernel(float* __restrict__ mx, float* __restrict__ den,
                            float* __restrict__ num)
{
    int i = blockIdx.x * blockDim.x + threadIdx.x;
    if (i < N_NODES * HEADS) { mx[i] = -3.0e38f; den[i] = 0.f; }
    if (i < N_NODES * F)     { num[i] = 0.f; }
}

// Sign-aware float atomic max via integer atomics (monotone bit tricks).
__device__ __forceinline__ void atomicMaxFloat(float* addr, float v)
{
    if (v >= 0.f) atomicMax((int*)addr, __float_as_int(v));
    else          atomicMin((unsigned int*)addr, __float_as_uint(v));
}

// ---------------------------------------------------------------------------
// Edge pass 1: segment max of leaky-relu logits per (dst, head)
// ---------------------------------------------------------------------------
__global__ void edge_max_kernel(const int* __restrict__ src,
                                const int* __restrict__ dst,
                                const float* __restrict__ a_s,
                                const float* __restrict__ a_d,
                                float* __restrict__ mx)
{
    int e = blockIdx.x * blockDim.x + threadIdx.x;
    if (e >= E_TOT) return;
    int s, d;
    if (e < N_EDGES) { s = src[e]; d = dst[e]; }
    else             { s = e - N_EDGES; d = s; }     // self loops
    #pragma unroll
    for (int hh = 0; hh < HEADS; ++hh) {
        float l = a_s[s * HEADS + hh] + a_d[d * HEADS + hh];
        l = (l > 0.f) ? l : NEG_SLOPE * l;
        atomicMaxFloat(&mx[d * HEADS + hh], l);
    }
}

// ---------------------------------------------------------------------------
// Edge pass 2: e = exp(logit - max); accumulate denom and UNNORMALIZED
// numerator sum_e e * h[src] (division deferred to finalize — identical math).
// ---------------------------------------------------------------------------
__global__ void edge_accum_kernel(const int* __restrict__ src,
                                  const int* __restrict__ dst,
                                  const float* __restrict__ a_s,
                                  const float* __restrict__ a_d,
                                  const float* __restrict__ mx,
                                  const float* __restrict__ h,
                                  float* __restrict__ den,
                                  float* __restrict__ num)
{
    int e = blockIdx.x * blockDim.x + threadIdx.x;
    if (e >= E_TOT) return;
    int s, d;
    if (e < N_EDGES) { s = src[e]; d = dst[e]; }
    else             { s = e - N_EDGES; d = s; }
    const float4* hp = (const float4*)(h + (size_t)s * F);   // 64B gather
    #pragma unroll
    for (int hh = 0; hh < HEADS; ++hh) {
        float l = a_s[s * HEADS + hh] + a_d[d * HEADS + hh];
        l = (l > 0.f) ? l : NEG_SLOPE * l;
        float w = __expf(l - mx[d * HEADS + hh]);
        atomicAdd(&den[d * HEADS + hh], w);
        float4 hv = hp[hh];
        float* nb = num + (size_t)d * F + hh * OUTC;
        atomicAdd(nb + 0, w * hv.x);
        atomicAdd(nb + 1, w * hv.y);
        atomicAdd(nb + 2, w * hv.z);
        atomicAdd(nb + 3, w * hv.w);
    }
}

// ---------------------------------------------------------------------------
// Finalize: out = num/den + bias; LayerNorm over 16 features; mean -> [N]
// ---------------------------------------------------------------------------
__global__ void finalize_kernel(const float* __restrict__ num,
                                const float* __restrict__ den,
                                const float* __restrict__ bias,
                                const float* __restrict__ gamma,
                                const float* __restrict__ beta,
                                float* __restrict__ out)
{
    int n = blockIdx.x * blockDim.x + threadIdx.x;
    if (n >= N_NODES) return;
    float v[F];
    float mu = 0.f;
    #pragma unroll
    for (int j = 0; j < F; ++j) {
        float t = num[(size_t)n * F + j] / den[n * HEADS + (j >> 2)] + bias[j];
        v[j] = t;
        mu += t;
    }
    mu *= (1.f / F);
    float var = 0.f;
    #pragma unroll
    for (int j = 0; j < F; ++j) { float c = v[j] - mu; var += c * c; }
    var *= (1.f / F);
    const float rs = rsqrtf(var + LN_EPS);
    float acc = 0.f;
    #pragma unroll
    for (int j = 0; j < F; ++j)
        acc += (v[j] - mu) * rs * gamma[j] + beta[j];
    out[n] = acc * (1.f / F);
}

// ---------------------------------------------------------------------------
extern "C" void kernel_launch(void* const* d_in, const int* in_sizes, int n_in,
                              void* d_out, int out_size, void* d_ws, size_t ws_size,
                              hipStream_t stream)
{
    (void)in_sizes; (void)n_in; (void)out_size; (void)ws_size;

    const float* x       = (const float*)d_in[0];
    const int*   ei      = (const int*)d_in[1];     // [2, E] (jax default int32)
    const float* W       = (const float*)d_in[2];
    const float* att_src = (const float*)d_in[3];
    const float* att_dst = (const float*)d_in[4];
    const float* bias    = (const float*)d_in[5];
    const float* gamma   = (const float*)d_in[6];
    const float* beta    = (const float*)d_in[7];
    float*       out     = (float*)d_out;

    const int* src = ei;             // edge_index[0]
    const int* dst = ei + N_EDGES;   // edge_index[1]

    // Workspace carve-up (floats): 4.8M floats = 19.2 MB
    float* ws  = (float*)d_ws;
    float* h   = ws;                                   // N*16
    float* a_s = h   + (size_t)N_NODES * F;            // N*4
    float* a_d = a_s + (size_t)N_NODES * HEADS;        // N*4
    float* mx  = a_d + (size_t)N_NODES * HEADS;        // N*4
    float* den = mx  + (size_t)N_NODES * HEADS;        // N*4
    float* num = den + (size_t)N_NODES * HEADS;        // N*16

    init_kernel<<<(N_NODES * F + 255) / 256, 256, 0, stream>>>(mx, den, num);
    gemm_att_kernel<<<(NTILES + 7) / 8, 256, 0, stream>>>(
        x, W, att_src, att_dst, h, a_s, a_d);
    edge_max_kernel<<<(E_TOT + 255) / 256, 256, 0, stream>>>(
        src, dst, a_s, a_d, mx);
    edge_accum_kernel<<<(E_TOT + 255) / 256, 256, 0, stream>>>(
        src, dst, a_s, a_d, mx, h, den, num);
    finalize_kernel<<<(N_NODES + 255) / 256, 256, 0, stream>>>(
        num, den, bias, gamma, beta, out);
}